// CapsuleLayer_18777597018146
// MI455X (gfx1250) — compile-verified
//
#include <hip/hip_runtime.h>
#include <hip/hip_bf16.h>

typedef float v2f __attribute__((ext_vector_type(2)));
typedef float v8f __attribute__((ext_vector_type(8)));

#define NIN   1152
#define NOUT  10
#define DIN   8
#define DOUT  16
#define NTILE 16               // 256 batch / 16 rows per WMMA tile
#define TILEF (16*NOUT*DOUT)   // 2560 floats per (tile,split) partial

// ---------------------------------------------------------------------------
// Pass kernel: one workgroup = (16-batch tile, Nin slice). Each wave builds
// u_hat 16x16 tiles with V_WMMA_F32_16X16X4_F32 and accumulates the routed
// sum s[b,j,e]. mode 1: uniform coupling 1/10. mode 2: c = softmax_j<u,v1>.
// mode 3: c = softmax_j<u, v1+v2>.
// ---------------------------------------------------------------------------
__global__ __launch_bounds__(256)
void caps_pass(const float* __restrict__ x, const float* __restrict__ W,
               const float* __restrict__ V1, const float* __restrict__ V2,
               float* __restrict__ P, int mode, int nsplit, int ipw)
{
    __shared__ float vt[TILEF];        // v (or v1+v2) tile for this batch tile
    __shared__ float sbuf[4][TILEF];   // cross-wave reduction buffers

    const int tid   = threadIdx.x;
    const int lane  = tid & 31;
    const int wave  = tid >> 5;        // 8 waves
    const int half  = lane >> 4;
    const int m     = lane & 15;       // N column / A row within half
    const int tile  = blockIdx.x;
    const int split = blockIdx.y;

    if (mode >= 2) {
        for (int k = tid; k < TILEF; k += 256) {   // 2560 = 10*256: uniform
            float v = V1[tile * TILEF + k];
            if (mode == 3) v += V2[tile * TILEF + k];
            vt[k] = v;
        }
    }
    __syncthreads();

    v8f s[NOUT];
    #pragma unroll
    for (int j = 0; j < NOUT; ++j)
        #pragma unroll
        for (int r = 0; r < 8; ++r) s[j][r] = 0.0f;

    const int bm = tile * 16 + m;      // batch row (A: M = lane&15)
    const int kb = half * 2;           // K pair base: lanes<16 -> {0,1}, else {2,3}
    const int i0 = split * ipw;
    const int i1 = (i0 + ipw < NIN) ? (i0 + ipw) : NIN;

    for (int i = i0 + wave; i < i1; i += 8) {      // wave-uniform: EXEC all 1s
        const float* xp = x + ((size_t)bm * NIN + i) * DIN;
        v2f alo, ahi;
        alo.x = xp[kb];     alo.y = xp[kb + 1];    // d = 0..3 half
        ahi.x = xp[kb + 4]; ahi.y = xp[kb + 5];    // d = 4..7 half

        v8f u[NOUT];
        const float* wp = W + (size_t)i * NOUT * DIN * DOUT;
        #pragma unroll
        for (int j = 0; j < NOUT; ++j) {
            const float* wj = wp + j * DIN * DOUT;
            v2f blo, bhi;                          // B rows K={kb,kb+1}, col m
            blo.x = wj[(kb + 0) * DOUT + m];
            blo.y = wj[(kb + 1) * DOUT + m];
            bhi.x = wj[(kb + 4) * DOUT + m];
            bhi.y = wj[(kb + 5) * DOUT + m];
            v8f c;
            #pragma unroll
            for (int r = 0; r < 8; ++r) c[r] = 0.0f;
            c = __builtin_amdgcn_wmma_f32_16x16x4_f32(false, alo, false, blo,
                                                      (short)0, c, false, false);
            c = __builtin_amdgcn_wmma_f32_16x16x4_f32(false, ahi, false, bhi,
                                                      (short)0, c, false, false);
            u[j] = c;   // u[j][r] = u_hat[b=tile*16+(r+8*half)][i][j][e=m]
        }

        if (mode == 1) {
            #pragma unroll
            for (int j = 0; j < NOUT; ++j) s[j] += 0.1f * u[j];
        } else {
            #pragma unroll
            for (int r = 0; r < 8; ++r) {
                const int M = r + 8 * half;        // batch row of this C slot
                float bl[NOUT];
                #pragma unroll
                for (int j = 0; j < NOUT; ++j) {
                    float t = u[j][r] * vt[(M * NOUT + j) * DOUT + m];
                    t += __shfl_xor(t, 1, 32);     // reduce over e (16 lanes/half)
                    t += __shfl_xor(t, 2, 32);
                    t += __shfl_xor(t, 4, 32);
                    t += __shfl_xor(t, 8, 32);
                    bl[j] = t;
                }
                float mx = bl[0];
                #pragma unroll
                for (int j = 1; j < NOUT; ++j) mx = fmaxf(mx, bl[j]);
                float ex[NOUT];
                float den = 0.0f;
                #pragma unroll
                for (int j = 0; j < NOUT; ++j) { ex[j] = __expf(bl[j] - mx); den += ex[j]; }
                const float inv = 1.0f / den;
                #pragma unroll
                for (int j = 0; j < NOUT; ++j) s[j][r] += (ex[j] * inv) * u[j][r];
            }
        }
    }

    // deterministic cross-wave reduction: waves 0-3 store, waves 4-7 add
    __syncthreads();
    if (wave < 4) {
        #pragma unroll
        for (int j = 0; j < NOUT; ++j)
            #pragma unroll
            for (int r = 0; r < 8; ++r)
                sbuf[wave][(j * 8 + r) * 32 + lane] = s[j][r];
    }
    __syncthreads();
    if (wave >= 4) {
        const int wb = wave - 4;
        #pragma unroll
        for (int j = 0; j < NOUT; ++j)
            #pragma unroll
            for (int r = 0; r < 8; ++r)
                sbuf[wb][(j * 8 + r) * 32 + lane] += s[j][r];
    }
    __syncthreads();

    float* outp = P + (size_t)(tile * nsplit + split) * TILEF;
    for (int k = tid; k < TILEF; k += 256) {       // uniform (2560 = 10*256)
        const int j   = k >> 8;
        const int rem = k & 255;
        const int r   = rem >> 5;
        const int l   = rem & 31;
        const int M   = r + 8 * (l >> 4);
        const int e   = l & 15;
        const float v = sbuf[0][k] + sbuf[1][k] + sbuf[2][k] + sbuf[3][k];
        outp[(M * NOUT + j) * DOUT + e] = v;       // [m][j][e] within chunk
    }
}

// ---------------------------------------------------------------------------
// Reduce split-partials (fixed order -> deterministic) and squash.
// One thread per (b, j) row; 2560 threads total.
// ---------------------------------------------------------------------------
__global__ __launch_bounds__(256)
void caps_squash(const float* __restrict__ P, float* __restrict__ Vout, int nsplit)
{
    const int t = blockIdx.x * 256 + threadIdx.x;  // 0..2559
    if (t >= 256 * NOUT) return;
    const int b = t / NOUT, j = t - b * NOUT;
    const int tile = b >> 4, m = b & 15;

    float acc[DOUT];
    #pragma unroll
    for (int e = 0; e < DOUT; ++e) acc[e] = 0.0f;
    for (int sp = 0; sp < nsplit; ++sp) {
        const float* p = P + (size_t)(tile * nsplit + sp) * TILEF + (m * NOUT + j) * DOUT;
        #pragma unroll
        for (int e = 0; e < DOUT; ++e) acc[e] += p[e];
    }
    float s2 = 0.0f;
    #pragma unroll
    for (int e = 0; e < DOUT; ++e) s2 += acc[e] * acc[e];
    const float scale = (s2 / (1.0f + s2)) * (1.0f / sqrtf(s2 + 1e-7f));
    float* o = Vout + (size_t)t * DOUT;            // [b][j][e] row-major
    #pragma unroll
    for (int e = 0; e < DOUT; ++e) o[e] = scale * acc[e];
}

// ---------------------------------------------------------------------------
extern "C" void kernel_launch(void* const* d_in, const int* in_sizes, int n_in,
                              void* d_out, int out_size, void* d_ws, size_t ws_size,
                              hipStream_t stream)
{
    const float* x = (const float*)d_in[0];   // [256,1152,8]
    const float* W = (const float*)d_in[1];   // [1152,10,8,16]
    float* out = (float*)d_out;               // [256,10,16]
    float* ws  = (float*)d_ws;

    const size_t vElems = (size_t)256 * NOUT * DOUT;   // 40960 floats per v buffer
    size_t availF = ws_size / sizeof(float);
    int nsplit = 16;
    while (nsplit > 1 &&
           ((size_t)NTILE * nsplit * TILEF + 2 * vElems) > availF)
        nsplit >>= 1;
    const int ipw = (NIN + nsplit - 1) / nsplit;

    float* P  = ws;                                     // partial s
    float* V1 = P + (size_t)NTILE * nsplit * TILEF;     // v after iter 1
    float* V2 = V1 + vElems;                            // v after iter 2

    dim3 g(NTILE, nsplit), blk(256);
    // iter 1: uniform coupling
    caps_pass<<<g, blk, 0, stream>>>(x, W, V1, V2, P, 1, nsplit, ipw);
    caps_squash<<<10, 256, 0, stream>>>(P, V1, nsplit);
    // iter 2: c = softmax(<u_hat, v1>)
    caps_pass<<<g, blk, 0, stream>>>(x, W, V1, V2, P, 2, nsplit, ipw);
    caps_squash<<<10, 256, 0, stream>>>(P, V2, nsplit);
    // iter 3: c = softmax(<u_hat, v1> + <u_hat, v2>)
    caps_pass<<<g, blk, 0, stream>>>(x, W, V1, V2, P, 3, nsplit, ipw);
    caps_squash<<<10, 256, 0, stream>>>(P, out, nsplit);
}